// Branch3d_stage0_69002944577835
// MI455X (gfx1250) — compile-verified
//
#include <hip/hip_runtime.h>
#include <hip/hip_bf16.h>

typedef float v2f __attribute__((ext_vector_type(2)));
typedef float v8f __attribute__((ext_vector_type(8)));

#define BN2 2
#define NPTS 4096
#define KNN 20
#define IH 480
#define IW 640
#define FH 240
#define FW 320
#define PIX (FH*FW)          // 76800
#define PIXB (BN2*PIX)       // 153600
#define FMOUT ((long)BN2*128*PIX) // 19660800 floats of fm in d_out

__device__ __forceinline__ v2f ld2(const float* p) { return *(const v2f*)p; }

// ---------------- generic fp32 WMMA GEMM: Y[R x Ncol] = A[R x K] * X[K x Ncol] ----------------
#define GF_XTRANS 1
#define GF_STORETRANS 2
#define GF_BIAS 4
#define GF_BN 8
#define GF_LRELU 16

template<int FLAGS, bool GUARDN>
__global__ __launch_bounds__(32) void gemm_wmma(
    const float* __restrict__ A, int ldA, long strideA,
    const float* __restrict__ X, int ldX, long strideX,
    float* __restrict__ Y, int ldY, long strideY,
    int R, int K, int Ncol,
    const float* __restrict__ bias, const float* __restrict__ bnp, int bnC)
{
  int lane = threadIdx.x & 31;
  int r0 = blockIdx.x * 16;
  int n0 = blockIdx.y * 64;
  int b  = blockIdx.z;
  A += (long)b * strideA; X += (long)b * strideX; Y += (long)b * strideY;
  int rowIn = lane & 15;
  int kk  = (lane >> 4) * 2;
  int col = lane & 15;

  v8f acc[4] = {};
  const float* ap = A + (long)(r0 + rowIn) * ldA + kk;   // contiguous pair (ldA even, kk even)

  const float* xp[4];
  bool cok[4];
#pragma unroll
  for (int t = 0; t < 4; ++t) {
    int c = n0 + t * 16 + col;
    cok[t] = !GUARDN || (c < Ncol);
    int cc = GUARDN ? (cok[t] ? c : 0) : c;
    if (FLAGS & GF_XTRANS) xp[t] = X + (long)cc * ldX + kk;
    else                   xp[t] = X + (long)kk * ldX + cc;
  }
  const long xstep = (FLAGS & GF_XTRANS) ? 4 : (long)4 * ldX;

#pragma unroll 2
  for (int k0 = 0; k0 < K; k0 += 4) {
    v2f a = ld2(ap);                 // single b64 load
    ap += 4;
#pragma unroll
    for (int t = 0; t < 4; ++t) {
      v2f bf;
      if (FLAGS & GF_XTRANS) {
        bf = ld2(xp[t]);             // contiguous pair: single b64 load
      } else if (GUARDN) {
        bf.x = cok[t] ? xp[t][0]   : 0.f;
        bf.y = cok[t] ? xp[t][ldX] : 0.f;
      } else {
        bf.x = xp[t][0];
        bf.y = xp[t][ldX];
      }
      xp[t] += xstep;
      acc[t] = __builtin_amdgcn_wmma_f32_16x16x4_f32(false, a, false, bf, (short)0, acc[t], false, false);
    }
  }

#pragma unroll
  for (int t = 0; t < 4; ++t) {
    int c = n0 + t * 16 + col;
    if (GUARDN && c >= Ncol) continue;
    float badd = 0.f, sc = 1.f, sh = 0.f;
    if (FLAGS & GF_BIAS) badd = bias[c];
    if (FLAGS & GF_BN) {
      float g = bnp[c], bb = bnp[bnC + c], mm = bnp[2 * bnC + c], vv = bnp[3 * bnC + c];
      sc = g * rsqrtf(vv + 1e-5f);
      sh = bb - mm * sc;
    }
#pragma unroll
    for (int i = 0; i < 8; ++i) {
      int m_ = i + ((lane >> 4) << 3);
      int rr = r0 + m_;
      float z = acc[t][i] + badd;
      z = z * sc + sh;
      if (FLAGS & GF_LRELU) z = z >= 0.f ? z : 0.2f * z;
      if (FLAGS & GF_STORETRANS) Y[(long)c * ldY + rr] = z;
      else                       Y[(long)rr * ldY + c] = z;
    }
  }
}

// ---------------- KNN: pairwise distance tiles via WMMA, then iterated argmax top-20 ----------------
__global__ __launch_bounds__(32) void knn_dist(
    const float* __restrict__ x,   // (C, N) for one batch
    const float* __restrict__ xx,  // (N)
    float* __restrict__ dist, int C, int N)
{
  int lane = threadIdx.x & 31;
  int r0 = blockIdx.x * 16, n0 = blockIdx.y * 64;
  int rowIn = lane & 15, kk = (lane >> 4) * 2, col = lane & 15;
  v8f acc[4] = {};
  const float* ap = x + (long)kk * N + r0 + rowIn;   // A[row][k] = x[k*N + row]
  const float* xp[4];
#pragma unroll
  for (int t = 0; t < 4; ++t) xp[t] = x + (long)kk * N + n0 + t * 16 + col;
  const long step = (long)4 * N;
#pragma unroll 2
  for (int k0 = 0; k0 < C; k0 += 4) {
    v2f a;
    a.x = ap[0];
    a.y = ap[N];
    ap += step;
#pragma unroll
    for (int t = 0; t < 4; ++t) {
      v2f bf;
      bf.x = xp[t][0];
      bf.y = xp[t][N];
      xp[t] += step;
      acc[t] = __builtin_amdgcn_wmma_f32_16x16x4_f32(false, a, false, bf, (short)0, acc[t], false, false);
    }
  }
#pragma unroll
  for (int t = 0; t < 4; ++t) {
    int c = n0 + t * 16 + col;
    float xc = xx[c];
#pragma unroll
    for (int i = 0; i < 8; ++i) {
      int m = i + ((lane >> 4) << 3);
      dist[(long)(r0 + m) * N + c] = 2.f * acc[t][i] - xx[r0 + m] - xc;
    }
  }
}

__global__ __launch_bounds__(32) void knn_select(float* dist, int* __restrict__ idxb, int N)
{
  int r = blockIdx.x;
  int lane = threadIdx.x;
  float* drow = dist + (long)r * N;
  for (int sel = 0; sel < KNN; ++sel) {
    float bv = -__builtin_inff(); int bi_ = 0x7fffffff;
    for (int c = lane; c < N; c += 32) {
      float v = drow[c];
      if (v > bv) { bv = v; bi_ = c; }
    }
#pragma unroll
    for (int off = 16; off > 0; off >>= 1) {
      float ov = __shfl_xor(bv, off, 32);
      int   oi = __shfl_xor(bi_, off, 32);
      if (ov > bv || (ov == bv && oi < bi_)) { bv = ov; bi_ = oi; }
    }
    if (lane == 0) { idxb[(long)r * KNN + sel] = bi_; drow[bi_] = -__builtin_inff(); }
    __syncthreads();
  }
}

// ---------------- image-side kernels ----------------
__global__ void preconv3x3(const float* __restrict__ in, const float* __restrict__ w,
                           const float* __restrict__ bias, float* __restrict__ out)
{
  long i = (long)blockIdx.x * 256 + threadIdx.x; // B*IH*IW*24
  if (i >= (long)BN2 * IH * IW * 24) return;
  int o = i % 24; long p = i / 24;
  int x = p % IW; p /= IW; int y = p % IH; int b = p / IH;
  float s = bias[o];
  for (int ic = 0; ic < 3; ++ic)
    for (int ky = 0; ky < 3; ++ky) {
      int yy = y + ky - 1; if (yy < 0 || yy >= IH) continue;
      for (int kx = 0; kx < 3; ++kx) {
        int xx_ = x + kx - 1; if (xx_ < 0 || xx_ >= IW) continue;
        s += w[((o * 3 + ic) * 3 + ky) * 3 + kx] * in[(((long)b * 3 + ic) * IH + yy) * IW + xx_];
      }
    }
  out[i] = s;
}

__global__ void ca_pool(const float* __restrict__ fs1, float* __restrict__ y)
{
  int i = blockIdx.x * 256 + threadIdx.x; // B*1120*24
  if (i >= BN2 * (IH + IW) * 24) return;
  int c = i % 24; int pos = (i / 24) % (IH + IW); int b = i / (24 * (IH + IW));
  float s = 0.f;
  if (pos < IH) {
    for (int xw = 0; xw < IW; ++xw) s += fs1[(((long)b * IH + pos) * IW + xw) * 24 + c];
    s /= (float)IW;
  } else {
    int xw = pos - IH;
    for (int h = 0; h < IH; ++h) s += fs1[(((long)b * IH + h) * IW + xw) * 24 + c];
    s /= (float)IH;
  }
  y[i] = s;
}

__global__ void ca_mlp(const float* __restrict__ y,
                       const float* __restrict__ c1w, const float* __restrict__ c1b,
                       const float* __restrict__ bnp,
                       const float* __restrict__ chw, const float* __restrict__ chb,
                       const float* __restrict__ cww, const float* __restrict__ cwb,
                       float* __restrict__ ah, float* __restrict__ aw)
{
  int i = blockIdx.x * 64 + threadIdx.x;
  if (i >= BN2 * (IH + IW)) return;
  int pos = i % (IH + IW), b = i / (IH + IW);
  float h[8];
  for (int j = 0; j < 8; ++j) {
    float s = c1b[j];
    for (int c = 0; c < 24; ++c) s += c1w[j * 24 + c] * y[(long)i * 24 + c];
    float g = bnp[j], bb = bnp[8 + j], mm = bnp[16 + j], vv = bnp[24 + j];
    s = (s - mm) * (g * rsqrtf(vv + 1e-5f)) + bb;
    float cl = s + 3.f; cl = cl < 0.f ? 0.f : (cl > 6.f ? 6.f : cl);
    h[j] = s * cl / 6.f;
  }
  if (pos < IH) {
    for (int o = 0; o < 24; ++o) {
      float s = chb[o];
      for (int j = 0; j < 8; ++j) s += chw[o * 8 + j] * h[j];
      ah[((long)b * IH + pos) * 24 + o] = 1.f / (1.f + expf(-s));
    }
  } else {
    int xw = pos - IH;
    for (int o = 0; o < 24; ++o) {
      float s = cwb[o];
      for (int j = 0; j < 8; ++j) s += cww[o * 8 + j] * h[j];
      aw[((long)b * IW + xw) * 24 + o] = 1.f / (1.f + expf(-s));
    }
  }
}

__global__ void ca_apply(float* __restrict__ fs1, const float* __restrict__ ah, const float* __restrict__ aw)
{
  long i = (long)blockIdx.x * 256 + threadIdx.x;
  if (i >= (long)BN2 * IH * IW * 24) return;
  int c = i % 24; long p = i / 24;
  int x = p % IW; p /= IW; int y = p % IH; int b = p / IH;
  fs1[i] *= ah[((long)b * IH + y) * 24 + c] * aw[((long)b * IW + x) * 24 + c];
}

// ---------------- point-side helper kernels ----------------
__global__ void compute_vu(const float* __restrict__ pc, int* __restrict__ vint, int* __restrict__ uib)
{
  int i = blockIdx.x * 256 + threadIdx.x;
  if (i >= BN2 * NPTS) return;
  int b = i / NPTS, n = i % NPTS;
  int v = (int)floorf(pc[((long)b * 8 + 0) * NPTS + n] + 240.f);
  int u = (int)floorf(pc[((long)b * 8 + 1) * NPTS + n] + 320.f);
  v = v < 0 ? 0 : (v > IH - 1 ? IH - 1 : v);
  u = u < 0 ? 0 : (u > IW - 1 ? IW - 1 : u);
  vint[i] = v; uib[i] = u;
}

__global__ void build_feat3d(const float* __restrict__ pc, const float* __restrict__ fs1,
                             const int* __restrict__ vint, const int* __restrict__ uib,
                             float* __restrict__ f3)
{
  long i = (long)blockIdx.x * 256 + threadIdx.x; // B*32*N
  if (i >= (long)BN2 * 32 * NPTS) return;
  int n = i % NPTS; long t = i / NPTS; int c = t % 32; int b = t / 32;
  float v;
  if (c < 8) v = pc[((long)b * 8 + c) * NPTS + n];
  else {
    int vv = vint[b * NPTS + n], uu = uib[b * NPTS + n];
    v = fs1[(((long)b * IH + vv) * IW + uu) * 24 + (c - 8)];
  }
  f3[i] = v;
}

__global__ void sum_sq(const float* __restrict__ x, float* __restrict__ xx, int C)
{
  int i = blockIdx.x * 256 + threadIdx.x;
  if (i >= BN2 * NPTS) return;
  int b = i / NPTS, n = i % NPTS;
  const float* xb = x + (long)b * C * NPTS + n;
  float s = 0.f;
  for (int c = 0; c < C; ++c) { float v = xb[(long)c * NPTS]; s += v * v; }
  xx[i] = s;
}

__global__ void sub_w(const float* __restrict__ W, float* __restrict__ Wd, int Cout, int Cin)
{
  int i = blockIdx.x * 256 + threadIdx.x;
  if (i >= Cout * Cin) return;
  int o = i / Cin, c = i % Cin;
  Wd[i] = W[o * 2 * Cin + Cin + c] - W[o * 2 * Cin + c];
}

__global__ void edge_max(const float* __restrict__ y1, const float* __restrict__ y2,
                         const int* __restrict__ idxb, const float* __restrict__ bnp,
                         float* __restrict__ out, int Cout)
{
  long i = (long)blockIdx.x * 256 + threadIdx.x; // B*Cout*N
  if (i >= (long)BN2 * Cout * NPTS) return;
  int n = i % NPTS; long t = i / NPTS; int o = t % Cout; int b = t / Cout;
  float g = bnp[o], bb = bnp[Cout + o], mm = bnp[2 * Cout + o], vv = bnp[3 * Cout + o];
  float sc = g * rsqrtf(vv + 1e-5f);
  float sh = bb - mm * sc;
  const float* y1r = y1 + ((long)b * Cout + o) * NPTS;
  float y2v = y2[((long)b * Cout + o) * NPTS + n];
  const int* id = idxb + ((long)b * NPTS + n) * KNN;
  float best = -__builtin_inff();
  for (int k = 0; k < KNN; ++k) {
    float z = y1r[id[k]] + y2v;
    z = z * sc + sh;
    z = z >= 0.f ? z : 0.2f * z;
    best = fmaxf(best, z);
  }
  out[((long)b * Cout + o) * NPTS + n] = best;
}

__global__ void build_xcat(const float* __restrict__ x1, const float* __restrict__ x2,
                           const float* __restrict__ x3, float* __restrict__ xcat)
{
  long i = (long)blockIdx.x * 256 + threadIdx.x; // B*N*256
  if (i >= (long)BN2 * NPTS * 256) return;
  int c = i % 256; long pn = i / 256; int n = pn % NPTS; int b = pn / NPTS;
  float v;
  if (c < 64)       v = x1[((long)b * 64 + c) * NPTS + n];
  else if (c < 128) v = x2[((long)b * 64 + (c - 64)) * NPTS + n];
  else              v = x3[((long)b * 128 + (c - 128)) * NPTS + n];
  xcat[i] = v;
}

// ---------------- fm-side kernels ----------------
__global__ void fill_f(float* __restrict__ p, float v, long n)
{
  long i = (long)blockIdx.x * 256 + threadIdx.x;
  if (i < n) p[i] = v;
}

__global__ void scatter_add(const float* __restrict__ pf, const int* __restrict__ vint,
                            const int* __restrict__ uib, float* __restrict__ fm)
{
  long i = (long)blockIdx.x * 256 + threadIdx.x; // B*N*64
  if (i >= (long)BN2 * NPTS * 64) return;
  int c = i & 63; long pn = i >> 6; int n = pn % NPTS; int b = pn / NPTS;
  int y = vint[b * NPTS + n] >> 1, x = uib[b * NPTS + n] >> 1;
  atomicAdd(&fm[(((long)b * FH + y) * FW + x) * 64 + c], pf[pn * 64 + c]);
}

__global__ void softmax64(const float* __restrict__ in, float* __restrict__ out)
{
  long p = (long)blockIdx.x * 256 + threadIdx.x;
  if (p >= (long)PIXB) return;
  const float* a = in + p * 64; float* o = out + p * 64;
  float mx = a[0];
  for (int c = 1; c < 64; ++c) mx = fmaxf(mx, a[c]);
  float s = 0.f;
  for (int c = 0; c < 64; ++c) { float e = expf(a[c] - mx); o[c] = e; s += e; }
  float inv = 1.f / s;
  for (int c = 0; c < 64; ++c) o[c] *= inv;
}

__global__ void dcn_sample(const float* __restrict__ value, const float* __restrict__ om,
                           float* __restrict__ samp)
{
  long i = (long)blockIdx.x * 256 + threadIdx.x; // PIXB*4
  if (i >= (long)PIXB * 4) return;
  int g = i & 3; long pg = i >> 2;
  long hw = pg % PIX; int b = pg / PIX;
  int h = hw / FW, w = hw % FW;
  float acc[16];
#pragma unroll
  for (int c = 0; c < 16; ++c) acc[c] = 0.f;
  const float* omp = om + pg * 108 + g * 27;
  const float* vb  = value + (long)b * PIX * 64 + g * 16;
  for (int kp = 0; kp < 9; ++kp) {
    float dx = omp[kp * 3 + 0], dy = omp[kp * 3 + 1], msk = omp[kp * 3 + 2];
    float lx = (float)w + (float)(kp % 3) - 1.f + dx;
    float ly = (float)h + (float)(kp / 3) - 1.f + dy;
    float x0f = floorf(lx), y0f = floorf(ly);
    float wx = lx - x0f, wy = ly - y0f;
    int x0 = (int)x0f, y0 = (int)y0f;
#pragma unroll
    for (int corner = 0; corner < 4; ++corner) {
      int xi = x0 + (corner & 1), yi = y0 + (corner >> 1);
      if (xi < 0 || xi >= FW || yi < 0 || yi >= FH) continue;
      float wgt = ((corner & 1) ? wx : 1.f - wx) * ((corner >> 1) ? wy : 1.f - wy);
      float mw = msk * wgt;
      const float* vp = vb + ((long)yi * FW + xi) * 64;
#pragma unroll
      for (int c = 0; c < 16; ++c) acc[c] += mw * vp[c];
    }
  }
  float* op = samp + pg * 64 + g * 16;
#pragma unroll
  for (int c = 0; c < 16; ++c) op[c] = acc[c];
}

__global__ void write_idx(const int* __restrict__ vint, const int* __restrict__ uib, float* __restrict__ out)
{
  int i = blockIdx.x * 256 + threadIdx.x;
  if (i >= BN2 * NPTS) return;
  out[i] = (float)(i / NPTS);
  out[BN2 * NPTS + i] = (float)(vint[i] >> 1);
  out[2 * BN2 * NPTS + i] = (float)(uib[i] >> 1);
}

// ---------------- host orchestration ----------------
template<int FLAGS, bool GUARDN = false>
static inline void launch_gemm(hipStream_t s,
                               const float* A, int ldA, long sA,
                               const float* X, int ldX, long sX,
                               float* Y, int ldY, long sY,
                               int R, int K, int Ncol, int batches,
                               const float* bias = nullptr, const float* bnp = nullptr, int bnC = 0)
{
  dim3 g((R + 15) / 16, (Ncol + 63) / 64, batches);
  gemm_wmma<FLAGS, GUARDN><<<g, 32, 0, s>>>(A, ldA, sA, X, ldX, sX, Y, ldY, sY, R, K, Ncol, bias, bnp, bnC);
}

extern "C" void kernel_launch(void* const* d_in, const int* in_sizes, int n_in,
                              void* d_out, int out_size, void* d_ws, size_t ws_size,
                              hipStream_t stream)
{
  (void)in_sizes; (void)n_in; (void)out_size;
  const float* pc       = (const float*)d_in[0];
  const float* feat_s00 = (const float*)d_in[1];
  const float* preconv_w = (const float*)d_in[2];
  const float* preconv_b = (const float*)d_in[3];
  const float* ca_c1_w = (const float*)d_in[4];
  const float* ca_c1_b = (const float*)d_in[5];
  const float* ca_bn   = (const float*)d_in[6];
  const float* ca_ch_w = (const float*)d_in[7];
  const float* ca_ch_b = (const float*)d_in[8];
  const float* ca_cw_w = (const float*)d_in[9];
  const float* ca_cw_b = (const float*)d_in[10];
  const float* conv1_w = (const float*)d_in[11];
  const float* bn1     = (const float*)d_in[12];
  const float* conv2_w = (const float*)d_in[13];
  const float* bn2     = (const float*)d_in[14];
  const float* conv3_w = (const float*)d_in[15];
  const float* bn3     = (const float*)d_in[16];
  const float* lin4a_w = (const float*)d_in[17];
  const float* lin4b_w = (const float*)d_in[18];
  const float* d5_off_w = (const float*)d_in[19];
  const float* d5_off_b = (const float*)d_in[20];
  const float* d5_val_w = (const float*)d_in[21];
  const float* d5_val_b = (const float*)d_in[22];
  const float* d5_out_w = (const float*)d_in[23];
  const float* d5_out_b = (const float*)d_in[24];
  const float* bn5      = (const float*)d_in[25];
  const float* d6_off_w = (const float*)d_in[26];
  const float* d6_off_b = (const float*)d_in[27];
  const float* d6_val_w = (const float*)d_in[28];
  const float* d6_val_b = (const float*)d_in[29];
  const float* d6_out_w = (const float*)d_in[30];
  const float* d6_out_b = (const float*)d_in[31];
  const float* bn6      = (const float*)d_in[32];
  const float* conv7a_w = (const float*)d_in[33];
  const float* conv7b_w = (const float*)d_in[34];

  char* ws = (char*)d_ws;
  size_t off = 0;
  auto take = [&](size_t bytes) { size_t r = off; off += (bytes + 255) & ~(size_t)255; return r; };

  size_t o_fs1 = take((size_t)BN2 * IH * IW * 24 * 4);  // also reused for xcat/tmp4/pfeat later
  size_t o_cay = take((size_t)BN2 * (IH + IW) * 24 * 4);
  size_t o_ah  = take((size_t)BN2 * IH * 24 * 4);
  size_t o_aw  = take((size_t)BN2 * IW * 24 * 4);
  size_t o_v   = take((size_t)BN2 * NPTS * 4);
  size_t o_u   = take((size_t)BN2 * NPTS * 4);
  size_t o_f3  = take((size_t)BN2 * 32 * NPTS * 4);
  size_t o_xx  = take((size_t)BN2 * NPTS * 4);
  size_t o_idx = take((size_t)BN2 * NPTS * KNN * 4);
  size_t o_y1  = take((size_t)BN2 * 128 * NPTS * 4);
  size_t o_y2  = take((size_t)BN2 * 128 * NPTS * 4);
  size_t o_wd  = take((size_t)128 * 128 * 4);
  size_t o_x1  = take((size_t)BN2 * 64 * NPTS * 4);
  size_t o_x2  = take((size_t)BN2 * 64 * NPTS * 4);
  size_t o_x3  = take((size_t)BN2 * 128 * NPTS * 4);
  size_t o_fmA = take((size_t)PIXB * 64 * 4);
  size_t o_fmB = take((size_t)PIXB * 64 * 4);
  size_t o_val = take((size_t)PIXB * 64 * 4);
  size_t o_om  = take((size_t)PIXB * 108 * 4);
  size_t o_smp = take((size_t)PIXB * 64 * 4);
  if (ws_size < off) return;

  // aliases (temporally disjoint reuse)
  size_t o_xcat = o_fs1;                          // B*N*256 f
  size_t o_tmp4 = o_fs1 + (size_t)BN2 * NPTS * 256 * 4;
  size_t o_pf   = o_tmp4 + (size_t)BN2 * NPTS * 256 * 4;
  size_t o_dist = o_fmA;                          // N*N f <= fmA+fmB

  float* fs1 = (float*)(ws + o_fs1);
  float* cay = (float*)(ws + o_cay);
  float* ahb = (float*)(ws + o_ah);
  float* awb = (float*)(ws + o_aw);
  int*   vint = (int*)(ws + o_v);
  int*   uib  = (int*)(ws + o_u);
  float* f3  = (float*)(ws + o_f3);
  float* xxb = (float*)(ws + o_xx);
  int*   idxb = (int*)(ws + o_idx);
  float* y1b = (float*)(ws + o_y1);
  float* y2b = (float*)(ws + o_y2);
  float* wdb = (float*)(ws + o_wd);
  float* x1b = (float*)(ws + o_x1);
  float* x2b = (float*)(ws + o_x2);
  float* x3b = (float*)(ws + o_x3);
  float* fmA = (float*)(ws + o_fmA);
  float* fmB = (float*)(ws + o_fmB);
  float* valb = (float*)(ws + o_val);
  float* omb  = (float*)(ws + o_om);
  float* smpb = (float*)(ws + o_smp);
  float* xcat = (float*)(ws + o_xcat);
  float* tmp4 = (float*)(ws + o_tmp4);
  float* pfb  = (float*)(ws + o_pf);
  float* distb = (float*)(ws + o_dist);
  float* outf = (float*)d_out;

  // ---- image branch ----
  {
    long n = (long)BN2 * IH * IW * 24;
    preconv3x3<<<dim3((n + 255) / 256), 256, 0, stream>>>(feat_s00, preconv_w, preconv_b, fs1);
    ca_pool<<<dim3((BN2 * (IH + IW) * 24 + 255) / 256), 256, 0, stream>>>(fs1, cay);
    ca_mlp<<<dim3((BN2 * (IH + IW) + 63) / 64), 64, 0, stream>>>(cay, ca_c1_w, ca_c1_b, ca_bn,
                                                                 ca_ch_w, ca_ch_b, ca_cw_w, ca_cw_b, ahb, awb);
    ca_apply<<<dim3((n + 255) / 256), 256, 0, stream>>>(fs1, ahb, awb);
  }
  compute_vu<<<dim3((BN2 * NPTS + 255) / 256), 256, 0, stream>>>(pc, vint, uib);
  build_feat3d<<<dim3((BN2 * 32 * NPTS + 255) / 256), 256, 0, stream>>>(pc, fs1, vint, uib, f3);

  // ---- edge-conv layers ----
  auto knn = [&](const float* x, int C) {
    sum_sq<<<dim3((BN2 * NPTS + 255) / 256), 256, 0, stream>>>(x, xxb, C);
    for (int b = 0; b < BN2; ++b) {
      knn_dist<<<dim3(NPTS / 16, NPTS / 64), 32, 0, stream>>>(x + (long)b * C * NPTS, xxb + (long)b * NPTS,
                                                              distb, C, NPTS);
      knn_select<<<dim3(NPTS), 32, 0, stream>>>(distb, idxb + (long)b * NPTS * KNN, NPTS);
    }
  };
  auto edge = [&](const float* x, int Cin, const float* W, int Cout, const float* bnp, float* xout) {
    knn(x, Cin);
    sub_w<<<dim3((Cout * Cin + 255) / 256), 256, 0, stream>>>(W, wdb, Cout, Cin);
    // y1 = W[:, :Cin] @ x   ;  y2 = (W[:, Cin:] - W[:, :Cin]) @ x
    launch_gemm<0>(stream, W,   2 * Cin, 0, x, NPTS, (long)Cin * NPTS, y1b, NPTS, (long)Cout * NPTS,
                   Cout, Cin, NPTS, BN2);
    launch_gemm<0>(stream, wdb, Cin,     0, x, NPTS, (long)Cin * NPTS, y2b, NPTS, (long)Cout * NPTS,
                   Cout, Cin, NPTS, BN2);
    edge_max<<<dim3(((long)BN2 * Cout * NPTS + 255) / 256), 256, 0, stream>>>(y1b, y2b, idxb, bnp, xout, Cout);
  };
  edge(f3,  32, conv1_w, 64,  bn1, x1b);
  edge(x1b, 64, conv2_w, 64,  bn2, x2b);
  edge(x2b, 64, conv3_w, 128, bn3, x3b);

  // ---- lin4: (B,N,256) @ (256,256) @ (256,64) ----
  build_xcat<<<dim3(((long)BN2 * NPTS * 256 + 255) / 256), 256, 0, stream>>>(x1b, x2b, x3b, xcat);
  launch_gemm<0>(stream, xcat, 256, (long)NPTS * 256, lin4a_w, 256, 0, tmp4, 256, (long)NPTS * 256,
                 NPTS, 256, 256, BN2);
  launch_gemm<0>(stream, tmp4, 256, (long)NPTS * 256, lin4b_w, 64, 0, pfb, 64, (long)NPTS * 64,
                 NPTS, 256, 64, BN2);

  // ---- scatter + softmax ----
  fill_f<<<dim3(((long)PIXB * 64 + 255) / 256), 256, 0, stream>>>(fmA, 0.f, (long)PIXB * 64);
  scatter_add<<<dim3(((long)BN2 * NPTS * 64 + 255) / 256), 256, 0, stream>>>(pfb, vint, uib, fmA);
  softmax64<<<dim3((PIXB + 255) / 256), 256, 0, stream>>>(fmA, fmB);

  // ---- DCNv4 layers (pixel-major, batch folded into rows) ----
  auto dcn = [&](const float* in, float* out,
                 const float* off_w, const float* off_b,
                 const float* val_w, const float* val_b,
                 const float* out_w, const float* out_b, const float* bnp) {
    launch_gemm<GF_BIAS>(stream, in, 64, 0, val_w, 64, 0, valb, 64, 0,
                         PIXB, 64, 64, 1, val_b);
    launch_gemm<GF_BIAS, true>(stream, in, 64, 0, off_w, 108, 0, omb, 108, 0,
                               PIXB, 64, 108, 1, off_b);
    dcn_sample<<<dim3(((long)PIXB * 4 + 255) / 256), 256, 0, stream>>>(valb, omb, smpb);
    launch_gemm<GF_BIAS | GF_BN | GF_LRELU>(stream, smpb, 64, 0, out_w, 64, 0, out, 64, 0,
                                            PIXB, 64, 64, 1, out_b, bnp, 64);
  };
  dcn(fmB, fmA, d5_off_w, d5_off_b, d5_val_w, d5_val_b, d5_out_w, d5_out_b, bn5);
  dcn(fmA, fmB, d6_off_w, d6_off_b, d6_val_w, d6_val_b, d6_out_w, d6_out_b, bn6);

  // ---- conv7a/b ----
  launch_gemm<GF_XTRANS>(stream, fmB, 64, 0, conv7a_w, 64, 0, valb, 64, 0,
                         PIXB, 64, 64, 1);
  // conv7b writes channel-first directly into d_out: per batch, Y[c*PIX + pix]
  launch_gemm<GF_XTRANS | GF_STORETRANS>(stream, valb, 64, (long)PIX * 64, conv7b_w, 64, 0, outf, PIX, (long)128 * PIX,
                                         PIX, 64, 128, BN2);

  write_idx<<<dim3((BN2 * NPTS + 255) / 256), 256, 0, stream>>>(vint, uib, outf + FMOUT);
}